// FingerprintEmbedding_19559281066709
// MI455X (gfx1250) — compile-verified
//
#include <hip/hip_runtime.h>
#include <hip/hip_bf16.h>

typedef __attribute__((ext_vector_type(2))) float v2f;
typedef __attribute__((ext_vector_type(8))) float v8f;

// Problem constants (fixed by the reference's setup_inputs)
#define B_  32
#define L_  4096
#define V_  10
#define F_  1024
#define D_  64

// -----------------------------------------------------------------------------
// Kernel 1: Pt[d][v] = sum_f fp_table[v][f] * W[d][f] + b[d]
// One workgroup, 4 waves. Wave w computes the 16-column tile d = w*16..w*16+15
// using V_WMMA_F32_16X16X4_F32 with M padded from V=10 to 16 (masked to zero).
//
// A (16x4, f32) per-lane layout (ISA 7.12.2): lanes 0-15 hold M=lane, K={k0,k0+1};
// lanes 16-31 hold M=lane-16, K={k0+2,k0+3}.  B (4x16) is symmetric with N=lane%16.
// D (16x16 f32): VGPR j holds M = j + 8*(lane>=16), N = lane%16.
// -----------------------------------------------------------------------------
__global__ void fp_proj_wmma(const float* __restrict__ fp_table,
                             const float* __restrict__ W,
                             const float* __restrict__ bias,
                             float* __restrict__ Pt /* [D_][16] */) {
    const int tid  = threadIdx.x;
    const int lane = tid & 31;
    const int wave = tid >> 5;              // 0..3 -> N tile of D
    const int m    = lane & 15;             // A-matrix row (v index, padded)
    const int kof  = (lane >> 4) << 1;      // 0 or 2: which K pair this half-wave holds
    const int n    = wave * 16 + (lane & 15); // output column d

    // Mask phantom rows v=10..15 to zero (and clamp the pointer so we never
    // read past the 10x1024 fp_table allocation).
    const float amask = (m < V_) ? 1.0f : 0.0f;
    const float* __restrict__ arow = fp_table + (m < V_ ? m : 0) * F_;
    const float* __restrict__ brow = W + n * F_;

    v8f acc = {};
#pragma unroll 8
    for (int k0 = 0; k0 < F_; k0 += 4) {
        v2f a, b;
        a.x = arow[k0 + kof + 0] * amask;
        a.y = arow[k0 + kof + 1] * amask;
        b.x = brow[k0 + kof + 0];
        b.y = brow[k0 + kof + 1];
        // 8 args: (neg_a, A, neg_b, B, c_mod, C, reuse_a, reuse_b)
        acc = __builtin_amdgcn_wmma_f32_16x16x4_f32(
            false, a, false, b, (short)0, acc, false, false);
    }

    const float bv    = bias[n];
    const int   rbase = (lane >> 4) << 3;   // 0 or 8: which M half this half-wave holds
#pragma unroll
    for (int j = 0; j < 8; ++j) {
        // Store TRANSPOSED: Pt[d=n][v=rbase+j]  (gather kernel wants Pt[d*16+v])
        Pt[n * 16 + rbase + j] = acc[j] + bv;
    }
}

// -----------------------------------------------------------------------------
// Kernel 2: out[(b*D + d)*L + l] = Pt[d*16 + indices[b*L + l]]
// Block = 256 threads; each block handles one b and a 1024-wide l chunk.
// Pt (4 KB) and the index chunk are staged in LDS once; each thread then emits
// 64 float4 stores (4 consecutive l's) -> fully coalesced b128 stores.
// Pt is stored transposed so the <=10 distinct LDS reads at a fixed d hit
// 10 distinct banks (conflict-free; equal indices broadcast).
// -----------------------------------------------------------------------------
__global__ void gather_out(const int* __restrict__ indices,
                           const float* __restrict__ Pt,
                           float* __restrict__ out) {
    __shared__ float Pt_s[D_ * 16];
    __shared__ int   idx_s[1024];

    const int tid = threadIdx.x;           // 0..255
    const int b   = blockIdx.y;            // 0..31
    const int l0  = blockIdx.x * 1024;     // 0,1024,2048,3072

#pragma unroll
    for (int i = tid; i < D_ * 16; i += 256) Pt_s[i] = Pt[i];

    const int* __restrict__ ip = indices + b * L_ + l0;
#pragma unroll
    for (int i = tid; i < 1024; i += 256) idx_s[i] = ip[i];

    __syncthreads();

    const int l  = tid << 2;               // 4 consecutive l per thread
    const int i0 = idx_s[l + 0];
    const int i1 = idx_s[l + 1];
    const int i2 = idx_s[l + 2];
    const int i3 = idx_s[l + 3];

    float* __restrict__ orow = out + (size_t)b * D_ * L_ + l0 + l;
#pragma unroll
    for (int d = 0; d < D_; ++d) {
        const float* p = Pt_s + d * 16;
        float4 v = make_float4(p[i0], p[i1], p[i2], p[i3]);
        *reinterpret_cast<float4*>(orow + (size_t)d * L_) = v;
    }
}

extern "C" void kernel_launch(void* const* d_in, const int* in_sizes, int n_in,
                              void* d_out, int out_size, void* d_ws, size_t ws_size,
                              hipStream_t stream) {
    const int*   indices  = (const int*)d_in[0];     // [B, L]
    const float* fp_table = (const float*)d_in[1];   // [V, F]
    const float* W        = (const float*)d_in[2];   // [D, F]
    const float* bias     = (const float*)d_in[3];   // [D]
    float*       out      = (float*)d_out;           // [1, B*D, L]
    float*       Pt       = (float*)d_ws;            // [D, 16] scratch (4 KB)

    // Stage 1: tiny WMMA GEMM -> Pt (transposed projected table)
    fp_proj_wmma<<<1, 128, 0, stream>>>(fp_table, W, bias, Pt);

    // Stage 2: bandwidth-bound gather/broadcast, 128 blocks
    dim3 grid(L_ / 1024, B_);
    gather_out<<<grid, 256, 0, stream>>>(indices, Pt, out);
}